// MimiEuclideanCodebook_69157563400884
// MI455X (gfx1250) — compile-verified
//
#include <hip/hip_runtime.h>
#include <hip/hip_bf16.h>

// MimiEuclideanCodebook for MI455X (gfx1250, wave32, WMMA).
//
// Shapes: x=[N=32768, D=256] f32, embed_sum=[K=2048, D=256], usage=[K].
// Dominant cost: dots = x @ embed^T  (34.4 GFLOP)  ->  V_WMMA_F32_16X16X4_F32.
// argmin_k (x^2 - 2 dot + e^2) == argmin_k (e^2 - 2 dot): drop the row constant.
// embed (2 MB) stays resident in the 192 MB L2; HBM traffic ~70 MB (~3 us at
// 23.3 TB/s), so the WMMA pipe is the bottleneck -> f32 matrix path for
// bit-faithful ordering vs the f32 reference.

typedef __attribute__((ext_vector_type(2))) float v2f;
typedef __attribute__((ext_vector_type(8))) float v8f;

constexpr int D_DIM = 256;          // feature dim (compile-time for codegen)
constexpr int ROWS_PER_BLOCK = 32;  // x rows staged in LDS per block
constexpr int MWAVES = 2;           // waves tiling M (16 rows each)
constexpr int KCOLS = 4;            // waves tiling K (16 clusters each)
constexpr int LDS_STRIDE = D_DIM + 4;  // +16B pad -> conflict-free A reads

// ---------------------------------------------------------------------------
// Kernel 1: embed = embed_sum / clamp(usage, 1e-5);  esq[k] = sum_d embed^2
// ---------------------------------------------------------------------------
__global__ __launch_bounds__(256) void vq_prep(const float* __restrict__ embed_sum,
                                               const float* __restrict__ usage,
                                               float* __restrict__ embed,
                                               float* __restrict__ esq) {
  const int k = blockIdx.x;
  const int t = threadIdx.x;  // t indexes d (D_DIM == blockDim.x == 256)
  float u = fmaxf(usage[k], 1e-5f);
  float v = embed_sum[(size_t)k * D_DIM + t] / u;
  embed[(size_t)k * D_DIM + t] = v;

  float s = v * v;
  #pragma unroll
  for (int off = 16; off; off >>= 1) s += __shfl_xor(s, off, 32);  // wave32
  __shared__ float red[8];
  if ((t & 31) == 0) red[t >> 5] = s;
  __syncthreads();
  if (t < 8) {
    float x = red[t];
    #pragma unroll
    for (int off = 4; off; off >>= 1) x += __shfl_xor(x, off, 32);
    if (t == 0) esq[k] = x;
  }
}

// ---------------------------------------------------------------------------
// Kernel 2: per-row argmin_k (esq[k] - 2 * <x_row, embed_k>) via f32 WMMA.
// Block = 256 threads = 8 waves = 2 (M) x 4 (K-col); block owns 32 x-rows.
// ---------------------------------------------------------------------------
__global__ __launch_bounds__(256) void vq_argmin(const float* __restrict__ x,
                                                 const float* __restrict__ embed,
                                                 const float* __restrict__ esq,
                                                 int* __restrict__ idx_out,
                                                 float* __restrict__ idx_out_f,
                                                 int K) {
  __shared__ float xt[ROWS_PER_BLOCK * LDS_STRIDE];
  __shared__ float redBest[KCOLS][ROWS_PER_BLOCK];
  __shared__ int   redIdx[KCOLS][ROWS_PER_BLOCK];

  const int t = threadIdx.x;
  const int rowBase = blockIdx.x * ROWS_PER_BLOCK;

  // Stage the 32x256 f32 x-tile into LDS (float4 / global_load_b128).
  {
    const float4* src = (const float4*)(x + (size_t)rowBase * D_DIM);
    #pragma unroll
    for (int i = 0; i < (ROWS_PER_BLOCK * D_DIM / 4) / 256; ++i) {
      int lin = i * 256 + t;          // float4 index within tile
      int r   = lin >> 6;             // / (D/4)
      int c4  = lin & 63;
      float4 v = src[lin];
      *(float4*)(&xt[r * LDS_STRIDE + c4 * 4]) = v;
    }
  }
  __syncthreads();

  const int wave   = t >> 5;
  const int lane   = t & 31;
  const int mwave  = wave & (MWAVES - 1);   // 0..1 -> rows [mwave*16, +16)
  const int kcol   = wave >> 1;             // 0..3 -> cluster column offset
  const int half   = lane >> 4;             // ISA: lanes 16-31 hold K+2 / M+8
  const int lane16 = lane & 15;

  // A fragment source row in LDS (lane16 == M within the 16-row tile).
  const float* aRow = &xt[(mwave * 16 + lane16) * LDS_STRIDE];

  float best[8];
  int   bidx[8];
  #pragma unroll
  for (int r = 0; r < 8; ++r) { best[r] = 3.4e38f; bidx[r] = 0; }

  for (int kk = kcol * 16; kk < K; kk += KCOLS * 16) {
    const int cluster = kk + lane16;  // B fragment: N == lane16
    const float* bRow = embed + (size_t)cluster * D_DIM;

    v8f acc = {0.f, 0.f, 0.f, 0.f, 0.f, 0.f, 0.f, 0.f};
    #pragma unroll 8
    for (int d = 0; d < D_DIM; d += 4) {
      const int dd = d + 2 * half;  // half 0 -> K{0,1}; half 1 -> K{2,3}
      v2f a = *(const v2f*)(aRow + dd);  // ds_load_b64, conflict-free (pad)
      v2f b = *(const v2f*)(bRow + dd);  // global_load_b64 (L2-resident)
      acc = __builtin_amdgcn_wmma_f32_16x16x4_f32(
          /*neg_a=*/false, a, /*neg_b=*/false, b,
          /*c_mod=*/(short)0, acc, /*reuse_a=*/false, /*reuse_b=*/false);
    }

    const float eq = esq[cluster];
    #pragma unroll
    for (int r = 0; r < 8; ++r) {
      // C/D layout: vgpr r, lane -> (m = half*8 + r, n = lane16)
      float dist = fmaf(-2.f, acc[r], eq);
      if (dist < best[r]) { best[r] = dist; bidx[r] = cluster; }  // kk ascends
    }
  }

  // Reduce over N (16 lanes in each half); tie -> smaller cluster index.
  #pragma unroll
  for (int r = 0; r < 8; ++r) {
    float b0 = best[r];
    int   i0 = bidx[r];
    #pragma unroll
    for (int off = 8; off; off >>= 1) {
      float ob = __shfl_xor(b0, off, 32);
      int   oi = __shfl_xor(i0, off, 32);
      if (ob < b0 || (ob == b0 && oi < i0)) { b0 = ob; i0 = oi; }
    }
    if (lane16 == 0) {
      int m = mwave * 16 + half * 8 + r;  // row within block tile
      redBest[kcol][m] = b0;
      redIdx[kcol][m]  = i0;
    }
  }
  __syncthreads();

  // Reduce the 4 K-column waves; write final index (int for gather, float out).
  if (t < ROWS_PER_BLOCK) {
    float b0 = redBest[0][t];
    int   i0 = redIdx[0][t];
    #pragma unroll
    for (int c = 1; c < KCOLS; ++c) {
      float ob = redBest[c][t];
      int   oi = redIdx[c][t];
      if (ob < b0 || (ob == b0 && oi < i0)) { b0 = ob; i0 = oi; }
    }
    idx_out[rowBase + t]   = i0;
    idx_out_f[rowBase + t] = (float)i0;
  }
}

// ---------------------------------------------------------------------------
// Kernel 3: quantize = embed[idx]  (4 rows per block, float4 copies)
// ---------------------------------------------------------------------------
__global__ __launch_bounds__(256) void vq_gather(const float* __restrict__ embed,
                                                 const int* __restrict__ idx,
                                                 float* __restrict__ out) {
  const int t   = threadIdx.x;
  const int row = blockIdx.x * 4 + (t >> 6);
  const int c4  = t & 63;
  const int k   = idx[row];
  float4 v = *(const float4*)(embed + (size_t)k * D_DIM + c4 * 4);
  *(float4*)(out + (size_t)row * D_DIM + c4 * 4) = v;
}

// ---------------------------------------------------------------------------
extern "C" void kernel_launch(void* const* d_in, const int* in_sizes, int n_in,
                              void* d_out, int out_size, void* d_ws, size_t ws_size,
                              hipStream_t stream) {
  const float* hidden    = (const float*)d_in[0];  // [N, D] f32
  const float* embed_sum = (const float*)d_in[1];  // [K, D] f32
  const float* usage     = (const float*)d_in[2];  // [K]    f32

  const int K = in_sizes[2];                 // 2048
  const int D = in_sizes[1] / K;             // 256 (== D_DIM)
  const int N = in_sizes[0] / D;             // 32768
  (void)n_in; (void)out_size;

  // Workspace layout: embed [K*D] f32 | esq [K] f32 | idx [N] i32  (~2.2 MB)
  float* ws_embed = (float*)d_ws;
  float* ws_esq   = ws_embed + (size_t)K * D;
  int*   ws_idx   = (int*)(ws_esq + K);
  (void)ws_size;

  float* out_q   = (float*)d_out;            // quantize: N*D floats
  float* out_idx = out_q + (size_t)N * D;    // indices as float values: N

  vq_prep<<<K, 256, 0, stream>>>(embed_sum, usage, ws_embed, ws_esq);
  vq_argmin<<<N / ROWS_PER_BLOCK, 256, 0, stream>>>(hidden, ws_embed, ws_esq,
                                                    ws_idx, out_idx, K);
  vq_gather<<<N / 4, 256, 0, stream>>>(ws_embed, ws_idx, out_q);
}